// MertForNERwithESD_74543452389463
// MI455X (gfx1250) — compile-verified
//
#include <hip/hip_runtime.h>
#include <hip/hip_bf16.h>
#include <stdint.h>

// ---------------------------------------------------------------------------
// MI455X (gfx1250) implementation.
//  - pack_bfrag_kernel: one-time repack of W_cls/W_esd into WMMA B-fragment
//    lane order (zero-padded to N=16): GEMM inner loop has no predication.
//  - fused_logits_kernel: two fp32 GEMMs via V_WMMA_F32_16X16X4_F32 (wave32).
//    A tiles stream global->LDS with GLOBAL_LOAD_ASYNC_TO_LDS_B128 (ASYNCcnt)
//    into a per-wave double buffer -> no VGPR staging array, no spill, and the
//    next chunk's DMA overlaps the current chunk's 16 WMMAs.
//  - crf_nll_kernel<K>: wave-per-sequence CRF forward scan, lane-per-tag,
//    shfl-based logsumexp; atomicAdd of -llh (weighted) into the loss scalar.
// Shapes: B=32, S=512, H=1024, NUM_TAGS=9, ESD_NUM_TAGS=5, RATIO=0.5
// ---------------------------------------------------------------------------

#define BATCH 32
#define SEQ   512
#define HID   1024
#define NTAG  9
#define NESD  5

typedef __attribute__((ext_vector_type(2))) float v2f;
typedef __attribute__((ext_vector_type(8))) float v8f;
typedef __attribute__((ext_vector_type(4))) int   i4;

#define ASTRIDE 68   // 64 + 4 pad floats: bank-conflict-free fragment reads

#if __has_builtin(__builtin_amdgcn_global_load_async_to_lds_b128)
#define HAVE_ASYNC_LDS 1
// src: global int4*, dst: LDS int4*, imm offset, imm cpol
#define GASYNC_B128(g, l)                                                     \
    __builtin_amdgcn_global_load_async_to_lds_b128(                           \
        (__attribute__((address_space(1))) i4*)(uintptr_t)(g),                \
        (__attribute__((address_space(3))) i4*)(l), 0, 0)
#else
#define HAVE_ASYNC_LDS 0
#endif

#if __has_builtin(__builtin_amdgcn_s_wait_asynccnt)
#define WAIT_ASYNC(n) __builtin_amdgcn_s_wait_asynccnt(n)
#else
#define WAIT_ASYNC(n) asm volatile("s_wait_asynccnt %0" ::"i"(n) : "memory")
#endif

// ---------------------------------------------------------------------------
// Repack W ([H,NT], row-major) into B-fragment order:
//   Bf[k4*32 + lane] = { Wpad[4*k4 + 2*h][n], Wpad[4*k4 + 2*h + 1][n] }
// with h = lane>>4, n = lane&15, pad cols (n >= NT) = 0.
// ---------------------------------------------------------------------------
__global__ __launch_bounds__(256) void pack_bfrag_kernel(
    const float* __restrict__ W_esd, const float* __restrict__ W_cls,
    float2* __restrict__ Bf_esd, float2* __restrict__ Bf_cls)
{
    const int tid  = blockIdx.x * 256 + threadIdx.x;   // [0, 16384)
    const int mat  = tid >> 13;
    const int rem  = tid & 8191;
    const int k4   = rem >> 5;
    const int lane = rem & 31;
    const int h    = lane >> 4;
    const int n    = lane & 15;
    const int k    = 4 * k4 + 2 * h;
    const float* __restrict__ W = mat ? W_cls : W_esd;
    const int NT = mat ? NTAG : NESD;
    float2 v;
    v.x = (n < NT) ? W[(size_t)k       * NT + n] : 0.0f;
    v.y = (n < NT) ? W[(size_t)(k + 1) * NT + n] : 0.0f;
    (mat ? Bf_cls : Bf_esd)[k4 * 32 + lane] = v;
}

__global__ __launch_bounds__(128) void fused_logits_kernel(
    const float* __restrict__ hidden,      // [B*S, H]
    const float* __restrict__ esd_hidden,  // [B*S, H]
    const float2* __restrict__ Bf_cls,     // [256*32] packed B frags
    const float* __restrict__ b_cls,       // [NTAG]
    const float2* __restrict__ Bf_esd,     // [256*32] packed B frags
    const float* __restrict__ b_esd,       // [NESD]
    const float* __restrict__ W_e2n,       // [B, NESD, NTAG]
    float* __restrict__ logits_out,        // [B*S, NTAG]
    float* __restrict__ esd_ws)            // [B*S, NESD]
{
    __shared__ float lds[4][2][16 * ASTRIDE];   // per-wave double buffer

    const int lane = threadIdx.x & 31;
    const int wave = threadIdx.x >> 5;
    const int tile = blockIdx.x * 4 + wave;    // 0..1023, 16 rows each
    const int rowBase = tile * 16;             // flat row index into [B*S]
    const int b = rowBase / SEQ;               // tiles never straddle batches
    const int half = lane >> 4;                // 0 | 1
    const int nn = lane & 15;                  // N column (pad cols >= NT -> 0)
    const int c4 = nn * 4;                     // staging column group

    v8f c_cls = {};
    v8f c_esd = {};

    // Two streaming GEMM passes: pass 0 = ESD, pass 1 = CLS.
    for (int which = 0; which < 2; ++which) {
        const float* __restrict__ X   = which ? hidden : esd_hidden;
        const float2* __restrict__ Bf = which ? Bf_cls : Bf_esd;
        v8f c = {};

#if HAVE_ASYNC_LDS
        // ---- issue async DMA for chunk 0 ----
        {
            float* dst = lds[wave][0];
            #pragma unroll
            for (int p = 0; p < 8; ++p) {
                const float* g = X + (size_t)(rowBase + p * 2 + half) * HID + c4;
                float* l = dst + (p * 2 + half) * ASTRIDE + c4;
                GASYNC_B128(g, l);
            }
        }
        for (int ci = 0; ci < HID / 64; ++ci) {
            const int kb = ci * 64;
            float* A = lds[wave][ci & 1];
            if (ci + 1 < HID / 64) {
                // ---- async DMA next chunk into other parity (overlaps WMMA) ----
                float* dst = lds[wave][(ci + 1) & 1];
                #pragma unroll
                for (int p = 0; p < 8; ++p) {
                    const float* g = X + (size_t)(rowBase + p * 2 + half) * HID
                                       + kb + 64 + c4;
                    float* l = dst + (p * 2 + half) * ASTRIDE + c4;
                    GASYNC_B128(g, l);
                }
                WAIT_ASYNC(8);      // current chunk done; next 8 still in flight
            } else {
                WAIT_ASYNC(0);
            }
            // ---- 16 x V_WMMA_F32_16X16X4_F32 over this chunk ----
            #pragma unroll
            for (int kk = 0; kk < 64; kk += 4) {
                const float2 av = *(const float2*)(A + nn * ASTRIDE + kk + 2 * half);
                const float2 bv = Bf[((kb + kk) >> 2) * 32 + lane];
                v2f a;  a.x = av.x;  a.y = av.y;
                v2f bb; bb.x = bv.x; bb.y = bv.y;
                c = __builtin_amdgcn_wmma_f32_16x16x4_f32(
                        false, a, false, bb, (short)0, c, false, false);
            }
        }
#else
        // ---- fallback: direct load -> ds_store staging (no long-lived regs) ----
        for (int kb = 0; kb < HID; kb += 64) {
            float* A = lds[wave][0];
            #pragma unroll
            for (int p = 0; p < 8; ++p) {
                const float4 v = *(const float4*)(X + (size_t)(rowBase + p * 2 + half) * HID
                                                  + kb + c4);
                *(float4*)(A + (p * 2 + half) * ASTRIDE + c4) = v;
            }
            #pragma unroll
            for (int kk = 0; kk < 64; kk += 4) {
                const float2 av = *(const float2*)(A + nn * ASTRIDE + kk + 2 * half);
                const float2 bv = Bf[((kb + kk) >> 2) * 32 + lane];
                v2f a;  a.x = av.x;  a.y = av.y;
                v2f bb; bb.x = bv.x; bb.y = bv.y;
                c = __builtin_amdgcn_wmma_f32_16x16x4_f32(
                        false, a, false, bb, (short)0, c, false, false);
            }
        }
#endif
        if (which) c_cls = c; else c_esd = c;
    }

    // ---- ESD bias, write esd_logits to workspace, exchange via LDS ----
    // C layout: reg r at lane l -> (m = r + 8*(l>>4), n = l&15)
    float* EX = lds[wave][0];                   // reuse staging region (idle)
    const float be = (nn < NESD) ? b_esd[nn] : 0.0f;
    #pragma unroll
    for (int r = 0; r < 8; ++r) {
        const int m = r + 8 * half;
        const float v = c_esd[r] + be;          // pad cols stay exactly 0
        EX[m * 17 + nn] = v;
        if (nn < NESD)
            esd_ws[(size_t)(rowBase + m) * NESD + nn] = v;
    }

    // ---- logits = cls + b_cls + esd @ W_e2n[b] ----
    const float bc = (nn < NTAG) ? b_cls[nn] : 0.0f;
    float we[NESD];
    #pragma unroll
    for (int e = 0; e < NESD; ++e)
        we[e] = (nn < NTAG) ? W_e2n[(size_t)b * NESD * NTAG + e * NTAG + nn] : 0.0f;

    #pragma unroll
    for (int r = 0; r < 8; ++r) {
        const int m = r + 8 * half;
        float acc = c_cls[r] + bc;
        #pragma unroll
        for (int e = 0; e < NESD; ++e)
            acc += EX[m * 17 + e] * we[e];      // same-address broadcast reads
        if (nn < NTAG)
            logits_out[(size_t)(rowBase + m) * NTAG + nn] = acc;
    }
}

// ---------------------------------------------------------------------------
// CRF negative log-likelihood: one wave (32 lanes) per sequence, lane = tag.
// Assumes attention mask is a contiguous prefix of ones (true for the inputs).
// Adds weight*(gold_score - logZ) into *loss (weight = -1 NER, -0.5 ESD).
// ---------------------------------------------------------------------------
template <int K>
__global__ __launch_bounds__(32) void crf_nll_kernel(
    const float* __restrict__ em,      // [B, S, K]
    const int*   __restrict__ tags,    // [B, S]
    const int*   __restrict__ mask,    // [B, S]
    const float* __restrict__ trans,   // [K, K]
    const float* __restrict__ start,   // [K]
    const float* __restrict__ endv,    // [K]
    float weight,
    float* __restrict__ loss)
{
    const int b    = blockIdx.x;
    const int lane = threadIdx.x;
    const float* __restrict__ emb = em + (size_t)b * SEQ * K;
    const int*   __restrict__ tgb = tags + b * SEQ;
    const int*   __restrict__ mkb = mask + b * SEQ;

    const bool act = lane < K;
    float Tcol[K];                         // trans[i][lane] (column for my tag)
    #pragma unroll
    for (int i = 0; i < K; ++i) Tcol[i] = act ? trans[i * K + lane] : 0.0f;

    // ---- numerator: gold-path score, parallel over time ----
    float part = 0.0f;
    int   cnt  = 0;
    for (int t = lane; t < SEQ; t += 32) {
        const int m = mkb[t];
        cnt += m;
        if (t >= 1 && m) {
            const int tg = tgb[t], pv = tgb[t - 1];
            part += trans[pv * K + tg] + emb[(size_t)t * K + tg];
        }
    }
    #pragma unroll
    for (int off = 16; off >= 1; off >>= 1) {
        part += __shfl_xor(part, off, 32);
        cnt  += __shfl_xor(cnt,  off, 32);
    }
    const int t0 = tgb[0];
    const float score = part + start[t0] + emb[t0] + endv[tgb[cnt - 1]];

    // ---- denominator: forward algorithm (online logsumexp over lanes) ----
    float alpha = act ? (start[lane] + emb[lane]) : -1e30f;
    for (int t = 1; t < SEQ; ++t) {
        const float e = act ? emb[(size_t)t * K + lane] : 0.0f;
        float mx = -1e30f, sm = 0.0f;
        #pragma unroll
        for (int i = 0; i < K; ++i) {
            const float ai = __shfl(alpha, i, 32);
            const float v  = ai + Tcol[i];
            const float nm = fmaxf(mx, v);
            sm = sm * __expf(mx - nm) + __expf(v - nm);
            mx = nm;
        }
        const float nxt = mx + __logf(sm) + e;
        if (mkb[t] && act) alpha = nxt;
    }
    const float v = act ? (alpha + endv[lane]) : -1e30f;
    float mx = v;
    #pragma unroll
    for (int off = 16; off >= 1; off >>= 1) mx = fmaxf(mx, __shfl_xor(mx, off, 32));
    float ex = __expf(v - mx);
    #pragma unroll
    for (int off = 16; off >= 1; off >>= 1) ex += __shfl_xor(ex, off, 32);
    const float logz = mx + __logf(ex);

    if (lane == 0) atomicAdd(loss, weight * (score - logz));
}

__global__ void init_loss_kernel(float* __restrict__ loss) { *loss = 0.0f; }

extern "C" void kernel_launch(void* const* d_in, const int* in_sizes, int n_in,
                              void* d_out, int out_size, void* d_ws, size_t ws_size,
                              hipStream_t stream) {
    (void)in_sizes; (void)n_in; (void)out_size; (void)ws_size;
    const float* hidden     = (const float*)d_in[0];
    const float* esd_hidden = (const float*)d_in[1];
    const int*   labels     = (const int*)  d_in[2];
    const int*   esd_labels = (const int*)  d_in[3];
    const int*   attn_mask  = (const int*)  d_in[4];
    const float* W_e2n      = (const float*)d_in[5];
    const float* W_cls      = (const float*)d_in[6];
    const float* b_cls      = (const float*)d_in[7];
    const float* W_esd      = (const float*)d_in[8];
    const float* b_esd      = (const float*)d_in[9];
    const float* trans      = (const float*)d_in[10];
    const float* start      = (const float*)d_in[11];
    const float* endv       = (const float*)d_in[12];
    const float* esd_trans  = (const float*)d_in[13];
    const float* esd_start  = (const float*)d_in[14];
    const float* esd_end    = (const float*)d_in[15];

    float* logits = (float*)d_out;                         // [B*S, NTAG]
    float* loss   = logits + (size_t)BATCH * SEQ * NTAG;   // scalar

    // workspace layout: esd_logits [B*S,NESD] | Bf_esd [8192] | Bf_cls [8192]
    float*  esd_ws = (float*)d_ws;
    float2* Bf_esd = (float2*)((char*)d_ws + (size_t)BATCH * SEQ * NESD * sizeof(float));
    float2* Bf_cls = Bf_esd + 256 * 32;

    init_loss_kernel<<<1, 1, 0, stream>>>(loss);

    pack_bfrag_kernel<<<64, 256, 0, stream>>>(W_esd, W_cls, Bf_esd, Bf_cls);

    // 1024 row tiles of 16, 4 waves/block
    fused_logits_kernel<<<256, 128, 0, stream>>>(
        hidden, esd_hidden, Bf_cls, b_cls, Bf_esd, b_esd, W_e2n, logits, esd_ws);

    crf_nll_kernel<NESD><<<BATCH, 32, 0, stream>>>(
        esd_ws, esd_labels, attn_mask, esd_trans, esd_start, esd_end, -0.5f, loss);

    crf_nll_kernel<NTAG><<<BATCH, 32, 0, stream>>>(
        logits, labels, attn_mask, trans, start, endv, -1.0f, loss);
}